// RecoverModel_1331439862201
// MI455X (gfx1250) — compile-verified
//
#include <hip/hip_runtime.h>

#define WAVES_PER_BLOCK 4
#define ROW_STRIDE 18   // pad 16-float rows to 18 words: conflict-free, 8B-aligned float2s

typedef float v2f __attribute__((ext_vector_type(2)));
typedef float v8f __attribute__((ext_vector_type(8)));

__device__ __forceinline__ void wave_lds_fence() {
  // same-wave LDS store->load ordering (CDNA5 split counters)
  asm volatile("s_wait_dscnt 0" ::: "memory");
}

__device__ __forceinline__ v8f vsplat8(float s) {
  v8f v;
#pragma unroll
  for (int j = 0; j < 8; ++j) v[j] = s;
  return v;
}

__device__ __forceinline__ v8f vrelu8(v8f v) {
#pragma unroll
  for (int j = 0; j < 8; ++j) v[j] = fmaxf(v[j], 0.0f);
  return v;
}

// D(16x16,f32) = A(16x16) * B(16x16) + C via 4 chained V_WMMA_F32_16X16X4_F32
__device__ __forceinline__ v8f wmma_16x16x16_f32(const v2f a[4], const v2f b[4], v8f c) {
#pragma unroll
  for (int k = 0; k < 4; ++k)
    c = __builtin_amdgcn_wmma_f32_16x16x4_f32(false, a[k], false, b[k], (short)0, c,
                                              false, false);
  return c;
}

// B-operand chunks for "X @ W^T": B[k,o] = W[o,k].
// Lane layout (32-bit B 4x16): lanes 0-15 hold K={0,1}, lanes 16-31 K={2,3}.
// => lane loads float2 from W[(lane&15)*16 + 4c + 2*(lane>>4)]
__device__ __forceinline__ void load_w_bfmt(const float* __restrict__ W, int lane, v2f b[4]) {
  const int o = lane & 15, hh = lane >> 4;
#pragma unroll
  for (int c = 0; c < 4; ++c)
    b[c] = *(const v2f*)(W + o * 16 + 4 * c + 2 * hh);
}

// store 16x16 C/D-layout accumulator to LDS rows [rowbase, rowbase+16)
// D layout: VGPR j, lane L -> row (j + 8*(L>>4)), col (L&15)
__device__ __forceinline__ void store_dlayout(float* Lp, int lane, int rowbase, v8f acc) {
  const int nn = lane & 15, hh = lane >> 4;
#pragma unroll
  for (int j = 0; j < 8; ++j)
    Lp[(rowbase + j + 8 * hh) * ROW_STRIDE + nn] = acc[j];
}

// load A-operand chunks from LDS; this lane's M-row lives at LDS row `row`.
// 32-bit A 16x4 layout: lane holds K = 4c + 2*(lane>>4) + {0,1}
__device__ __forceinline__ void load_afmt(const float* Lp, int lane, int row, v2f a[4]) {
  const int hh = lane >> 4;
#pragma unroll
  for (int c = 0; c < 4; ++c)
    a[c] = *(const v2f*)(Lp + row * ROW_STRIDE + 4 * c + 2 * hh);
}

__global__ __launch_bounds__(WAVES_PER_BLOCK * 32) void recover_model_kernel(
    const int* __restrict__ x, const float* __restrict__ phenos,
    const float* __restrict__ emb,
    const float* __restrict__ W1, const float* __restrict__ b1,
    const float* __restrict__ W2, const float* __restrict__ b2,
    const float* __restrict__ W3, const float* __restrict__ b3,
    const float* __restrict__ BW, const float* __restrict__ BD,
    const float* __restrict__ Boff,
    const float* __restrict__ Wc1, const float* __restrict__ bc1,
    const float* __restrict__ Wc2, const float* __restrict__ bc2,
    float* __restrict__ out, int ntiles) {
  __shared__ float lds[WAVES_PER_BLOCK][32 * ROW_STRIDE];
  const int lane = threadIdx.x & 31;
  const int wave = threadIdx.x >> 5;
  const int tile = blockIdx.x * WAVES_PER_BLOCK + wave;  // 16 samples per tile
  if (tile >= ntiles) return;                            // wave-uniform; EXEC stays full
  float* Lp = lds[wave];
  const int nn = lane & 15;  // lane's column / sample index
  const int hh = lane >> 4;

  // hoist all weight B-operands (shared across every tile)
  v2f w1b[4], w2b[4], w3b[4], wc1b[4];
  load_w_bfmt(W1, lane, w1b);
  load_w_bfmt(W2, lane, w2b);
  load_w_bfmt(W3, lane, w3b);
  load_w_bfmt(Wc1, lane, wc1b);
  const float bias1 = b1[nn], bias2 = b2[nn], bias3 = b3[nn];

  // ---- 3-layer MLP over 32 gathered rows (two 16-row WMMA tiles) ----
#pragma unroll
  for (int t = 0; t < 2; ++t) {
    const int rb = t * 16;                  // LDS staging rows for this half
    const int row = tile * 32 + rb + nn;    // flat gathered-row index (h = emb[x].reshape)
    const int g = x[row];
    const float* er = emb + (long)g * 16;
    v2f a[4];
#pragma unroll
    for (int c = 0; c < 4; ++c)             // A-operand straight from the gather
      a[c] = *(const v2f*)(er + 4 * c + 2 * hh);

    v8f acc = vsplat8(bias1);
    acc = wmma_16x16x16_f32(a, w1b, acc);
    acc = vrelu8(acc);
    store_dlayout(Lp, lane, rb, acc);       // D-layout -> A-layout via LDS
    wave_lds_fence();
    load_afmt(Lp, lane, rb + nn, a);

    acc = vsplat8(bias2);
    acc = wmma_16x16x16_f32(a, w2b, acc);
    acc = vrelu8(acc);
    store_dlayout(Lp, lane, rb, acc);
    wave_lds_fence();
    load_afmt(Lp, lane, rb + nn, a);

    acc = vsplat8(bias3);
    acc = wmma_16x16x16_f32(a, w3b, acc);   // no relu on layer 3
    store_dlayout(Lp, lane, rb, acc);       // final e-vectors for rows rb..rb+15
    wave_lds_fence();
  }

  // ---- split even/odd rows into per-sample e0/e1 A-operands ----
  v2f e0[4], e1[4];
  load_afmt(Lp, lane, 2 * nn, e0);      // sample nn, slot 0
  load_afmt(Lp, lane, 2 * nn + 1, e1);  // sample nn, slot 1
  wave_lds_fence();                     // loads retired before LDS reuse below

  // ---- bilinear: z[s,n] = sum_m A[s,m,n]*C[s,m,n]*BD[n,m] + Boff[n] ----
  // chunk m: A-tile = E0 @ M1^T cols (m,0..15); B[k,n] = BW[n,m,k]
  //          C-tile = E1 @ M2^T cols (m,0..15); B[k,n] = BW[m,n,k]
  v8f z = vsplat8(Boff[nn]);
  for (int m = 0; m < 16; ++m) {
    v2f bA[4], bC[4];
#pragma unroll
    for (int c = 0; c < 4; ++c) {
      const int kk = 4 * c + 2 * hh;
      bA[c] = *(const v2f*)(BW + nn * 256 + m * 16 + kk);  // BW[n][m][k]
      bC[c] = *(const v2f*)(BW + m * 256 + nn * 16 + kk);  // BW[m][n][k]
    }
    v8f At = vsplat8(0.0f);
    At = wmma_16x16x16_f32(e0, bA, At);
    v8f Ct = vsplat8(0.0f);
    Ct = wmma_16x16x16_f32(e1, bC, Ct);
    const float bd = BD[nn * 16 + m];   // BD[n,m]; n == lane's column in D layout
#pragma unroll
    for (int j = 0; j < 8; ++j)
      z[j] += At[j] * Ct[j] * bd;
  }

  // ---- head: z2 = relu(z @ Wc1^T + bc1) ----
  store_dlayout(Lp, lane, 0, z);
  wave_lds_fence();
  v2f az[4];
  load_afmt(Lp, lane, nn, az);
  v8f z2 = vsplat8(bc1[nn]);
  z2 = wmma_16x16x16_f32(az, wc1b, z2);
  z2 = vrelu8(z2);

  // ---- final 16->1 projection + phenos sum ----
  const float wc2 = Wc2[nn];
  const float cbias = bc2[0];
#pragma unroll
  for (int j = 0; j < 8; ++j) {
    float v = z2[j] * wc2;          // z2[s, n] * Wc2[n], n = lane&15
    v += __shfl_xor(v, 1);          // reduce each 16-lane half (wave32)
    v += __shfl_xor(v, 2);
    v += __shfl_xor(v, 4);
    v += __shfl_xor(v, 8);
    if (nn == 0) {                  // lanes 0 and 16 hold the two half-sums
      const int s = j + 8 * hh;
      const int gb = tile * 16 + s;
      out[gb] = phenos[2 * gb] + phenos[2 * gb + 1] + v + cbias;
    }
  }
}

extern "C" void kernel_launch(void* const* d_in, const int* in_sizes, int n_in,
                              void* d_out, int out_size, void* d_ws, size_t ws_size,
                              hipStream_t stream) {
  const int*   x      = (const int*)d_in[0];
  const float* phenos = (const float*)d_in[1];
  const float* emb    = (const float*)d_in[2];
  const float* W1     = (const float*)d_in[3];
  const float* b1     = (const float*)d_in[4];
  const float* W2     = (const float*)d_in[5];
  const float* b2     = (const float*)d_in[6];
  const float* W3     = (const float*)d_in[7];
  const float* b3     = (const float*)d_in[8];
  const float* BW     = (const float*)d_in[9];
  const float* BD     = (const float*)d_in[10];
  const float* Boff   = (const float*)d_in[11];
  const float* Wc1    = (const float*)d_in[12];
  const float* bc1    = (const float*)d_in[13];
  const float* Wc2    = (const float*)d_in[14];
  const float* bc2    = (const float*)d_in[15];
  float* out = (float*)d_out;

  const int Btot   = out_size;              // one output per sample
  const int ntiles = (Btot + 15) / 16;      // 16 samples per wave
  const int blocks = (ntiles + WAVES_PER_BLOCK - 1) / WAVES_PER_BLOCK;

  recover_model_kernel<<<blocks, WAVES_PER_BLOCK * 32, 0, stream>>>(
      x, phenos, emb, W1, b1, W2, b2, W3, b3, BW, BD, Boff, Wc1, bc1, Wc2, bc2,
      out, ntiles);
}